// _PointcloudRenderingModel_69810398429722
// MI455X (gfx1250) — compile-verified
//
#include <hip/hip_runtime.h>
#include <hip/hip_bf16.h>

// ---------------- problem constants ----------------
#define H_      256
#define W_      256
#define KTOP    10
#define P_      20000
#define RAD_    0.015f
#define R2_     (RAD_ * RAD_)
#define ZFAR_   1.0e10f
#define EPS_    1.0e-4f
#define NTX_    16          // tiles in x (16px tiles)
#define NTY_    16
#define RPIX_   (RAD_ * (W_ * 0.5f))   // 1.92 pixels

#define CHUNK_  128         // candidates staged to LDS per barrier
#define LG_CHUNK_ 7

// workspace layout (bytes)
#define WS_PROJ_    0
#define WS_COUNTS_  320000
#define WS_OFFS_    321024
#define WS_CURSOR_  322048
#define WS_LIST_    323072
#define WS_DUMMY_   643072      // 16-byte dummy record; byte offset from proj base == WS_DUMMY_

typedef __attribute__((ext_vector_type(2))) float v2f;
typedef __attribute__((ext_vector_type(8))) float v8f;

// ---------------- K1: project + per-tile count ----------------
__global__ __launch_bounds__(256)
void proj_count_kernel(const float* __restrict__ pts,
                       const float* __restrict__ Rm,
                       const float* __restrict__ Tv,
                       const float* __restrict__ focal,
                       const float* __restrict__ principal,
                       float4* __restrict__ proj,
                       int* __restrict__ counts)
{
    int i = blockIdx.x * blockDim.x + threadIdx.x;
    if (i >= P_) return;
    float px = pts[3*i+0], py = pts[3*i+1], pz = pts[3*i+2];
    // pc = p @ R + T   (pc[j] = sum_k p[k] * R[k][j])
    float c0 = px*Rm[0] + py*Rm[3] + pz*Rm[6] + Tv[0];
    float c1 = px*Rm[1] + py*Rm[4] + pz*Rm[7] + Tv[1];
    float c2 = px*Rm[2] + py*Rm[5] + pz*Rm[8] + Tv[2];
    float z    = c2;
    float invz = 1.0f / z;
    float xn = focal[0]*c0*invz + principal[0];
    float yn = focal[1]*c1*invz + principal[1];
    float4 rec;
    rec.x = xn; rec.y = yn; rec.z = z; rec.w = __int_as_float(i);
    proj[i] = rec;
    if (!(z > 0.01f)) return;                     // invalid -> never covered
    // pixel-space center: xs[i] - xn = 2*(fx - i)/W  =>  |fx - i| < RPIX
    float fx = (1.0f - xn) * (W_*0.5f) - 0.5f;
    float fy = (1.0f - yn) * (H_*0.5f) - 0.5f;
    float xlo = ceilf (fx - RPIX_), xhi = floorf(fx + RPIX_);
    float ylo = ceilf (fy - RPIX_), yhi = floorf(fy + RPIX_);
    if (xhi < 0.0f || xlo > (float)(W_-1) || yhi < 0.0f || ylo > (float)(H_-1)) return;
    int ix0 = (int)fmaxf(xlo, 0.0f), ix1 = (int)fminf(xhi, (float)(W_-1));
    int iy0 = (int)fmaxf(ylo, 0.0f), iy1 = (int)fminf(yhi, (float)(H_-1));
    int tx0 = ix0 >> 4, tx1 = ix1 >> 4;
    int ty0 = iy0 >> 4, ty1 = iy1 >> 4;
    for (int ty = ty0; ty <= ty1; ++ty)
        for (int tx = tx0; tx <= tx1; ++tx)
            atomicAdd(&counts[ty*NTX_ + tx], 1);
}

// ---------------- K2: exclusive scan over 256 tile counts + dummy record ------
__global__ __launch_bounds__(256)
void scan_kernel(const int* __restrict__ counts,
                 int* __restrict__ offs,
                 int* __restrict__ cursor,
                 float4* __restrict__ dummy)
{
    __shared__ int sm[256];
    int t = threadIdx.x;
    int v = counts[t];
    sm[t] = v;
    __syncthreads();
    #pragma unroll
    for (int off = 1; off < 256; off <<= 1) {
        int add = (t >= off) ? sm[t - off] : 0;
        __syncthreads();
        sm[t] += add;
        __syncthreads();
    }
    int excl = sm[t] - v;
    offs[t]   = excl;
    cursor[t] = excl;
    if (t == 0) {
        float4 d;
        d.x = 1.0e4f; d.y = 1.0e4f; d.z = ZFAR_; d.w = __int_as_float(0);
        *dummy = d;       // pad record: never passes the d^2 < R^2 prune
    }
}

// ---------------- K3: scatter point indices into per-tile lists ----------------
__global__ __launch_bounds__(256)
void scatter_kernel(const float4* __restrict__ proj,
                    int* __restrict__ cursor,
                    unsigned* __restrict__ list)
{
    int i = blockIdx.x * blockDim.x + threadIdx.x;
    if (i >= P_) return;
    float4 rec = proj[i];
    float z = rec.z;
    if (!(z > 0.01f)) return;
    float fx = (1.0f - rec.x) * (W_*0.5f) - 0.5f;
    float fy = (1.0f - rec.y) * (H_*0.5f) - 0.5f;
    float xlo = ceilf (fx - RPIX_), xhi = floorf(fx + RPIX_);
    float ylo = ceilf (fy - RPIX_), yhi = floorf(fy + RPIX_);
    if (xhi < 0.0f || xlo > (float)(W_-1) || yhi < 0.0f || ylo > (float)(H_-1)) return;
    int ix0 = (int)fmaxf(xlo, 0.0f), ix1 = (int)fminf(xhi, (float)(W_-1));
    int iy0 = (int)fmaxf(ylo, 0.0f), iy1 = (int)fminf(yhi, (float)(H_-1));
    int tx0 = ix0 >> 4, tx1 = ix1 >> 4;
    int ty0 = iy0 >> 4, ty1 = iy1 >> 4;
    for (int ty = ty0; ty <= ty1; ++ty)
        for (int tx = tx0; tx <= tx1; ++tx) {
            int pos = atomicAdd(&cursor[ty*NTX_ + tx], 1);
            list[pos] = (unsigned)i;
        }
}

// ---------------- K4: per-tile render --------------------------------------
// Async-LDS double-buffered candidate staging + WMMA distance pruning +
// per-lane top-K insertion + half-wave merge + front-to-back compositing.
__global__ __launch_bounds__(512)
void render_kernel(const float4* __restrict__ proj,
                   const float* __restrict__ feats,
                   const int* __restrict__ offs,
                   const int* __restrict__ counts,
                   const unsigned* __restrict__ list,
                   float* __restrict__ out)
{
    __shared__ float4 smrec[2][CHUNK_];

    const int tile = blockIdx.x;
    const int tx = tile & (NTX_-1), ty = tile >> 4;
    const int tid  = threadIdx.x;
    const int lane = tid & 31;
    const int wave = tid >> 5;            // pixel row within tile (16 waves)
    const int col  = lane & 15;           // pixel col within tile
    const bool hi  = (lane >= 16);

    const int pi = tx*16 + col;           // global pixel column i
    const int pj = ty*16 + wave;          // global pixel row j
    const float xpx = 1.0f - (2.0f*pi + 1.0f) * (1.0f/(float)W_);
    const float ypx = 1.0f - (2.0f*pj + 1.0f) * (1.0f/(float)H_);

    // B operand (pixels, 4x16): column n = pixel. Rows: [-2x, -2y, 1, x^2+y^2]
    v2f bmat;
    bmat.x = hi ? 1.0f                   : (-2.0f * xpx);
    bmat.y = hi ? (xpx*xpx + ypx*ypx)    : (-2.0f * ypx);

    // per-lane top-K sorted ascending by z
    float zt[KTOP], dt[KTOP];
    int   it[KTOP];
    #pragma unroll
    for (int s = 0; s < KTOP; ++s) { zt[s] = ZFAR_; dt[s] = 0.0f; it[s] = 0; }

    const int off = offs[tile];
    const int n   = counts[tile];

    const unsigned long long projbase = (unsigned long long)proj;
    const unsigned lds0 = (unsigned)(unsigned long long)&smrec[0][0];

    // issue an async gather of one chunk of candidate records into LDS buffer
    auto stage_issue = [&](int chbase, int buf) {
        if (tid < CHUNK_) {
            int c = chbase + tid;
            unsigned voff;
            if (c < n) {
                unsigned pidx = list[off + c];
                voff = pidx * 16u;                       // byte offset of proj[pidx]
            } else {
                voff = (unsigned)(WS_DUMMY_ - WS_PROJ_); // pad record
            }
            unsigned loff = lds0 + (unsigned)buf * (CHUNK_ * 16u) + (unsigned)tid * 16u;
            asm volatile("global_load_async_to_lds_b128 %0, %1, %2"
                         :: "v"(loff), "v"(voff), "s"(projbase)
                         : "memory");
        }
    };

    const int nchunks = (n + CHUNK_ - 1) >> LG_CHUNK_;
    for (int ch = 0; ch < nchunks; ++ch) {
        const int buf = ch & 1;
        if (ch == 0) {                                   // prologue stage
            stage_issue(0, 0);
            asm volatile("s_wait_asynccnt 0x0" ::: "memory");
            __syncthreads();
        }
        if (ch + 1 < nchunks)                            // overlap next gather
            stage_issue((ch + 1) << LG_CHUNK_, buf ^ 1);

        int rem = n - (ch << LG_CHUNK_); if (rem > CHUNK_) rem = CHUNK_;
        const int ngroups = (rem + 15) >> 4;
        const float4* sbuf = smrec[buf];

        for (int g = 0; g < ngroups; ++g) {
            // A operand (candidates, 16x4): row m = [x_p, y_p, x_p^2+y_p^2, 1]
            float4 rec = sbuf[g*16 + col];
            v2f amat;
            amat.x = hi ? (rec.x*rec.x + rec.y*rec.y) : rec.x;
            amat.y = hi ? 1.0f                        : rec.y;

            v8f cz8 = {};
            // D[m][n] = -2 x_p x_px - 2 y_p y_px + (x_p^2+y_p^2) + (x_px^2+y_px^2) = d^2
            v8f d = __builtin_amdgcn_wmma_f32_16x16x4_f32(
                        false, amat, false, bmat, (short)0, cz8, false, false);

            const int mbase = g*16 + (hi ? 8 : 0);
            #pragma unroll
            for (int r = 0; r < 8; ++r) {
                float d2a = d[r];
                if (d2a < R2_ * 1.01f) {            // WMMA prune with slack
                    float4 rc = sbuf[mbase + r];    // one ds_load_b128
                    float dx = xpx - rc.x, dy = ypx - rc.y;
                    float d2 = dx*dx + dy*dy;       // exact (matches ref rounding)
                    if (d2 < R2_) {
                        float cz = rc.z, cd = d2;
                        int   ci = __float_as_int(rc.w);
                        #pragma unroll
                        for (int s = 0; s < KTOP; ++s) {   // bubble insert (swap chain)
                            bool lt = cz < zt[s];
                            float tz = lt ? zt[s] : cz;
                            float td = lt ? dt[s] : cd;
                            int   ti = lt ? it[s] : ci;
                            zt[s] = lt ? cz : zt[s];
                            dt[s] = lt ? cd : dt[s];
                            it[s] = lt ? ci : it[s];
                            cz = tz; cd = td; ci = ti;
                        }
                    }
                }
            }
        }
        asm volatile("s_wait_asynccnt 0x0" ::: "memory");  // next chunk landed
        __syncthreads();
    }

    // ---- merge the two half-wave lists for the same pixel (lane <-> lane^16) ----
    float rz[KTOP], rd[KTOP];
    int   ri[KTOP];
    #pragma unroll
    for (int s = 0; s < KTOP; ++s) {
        rz[s] = __shfl_xor(zt[s], 16, 32);
        rd[s] = __shfl_xor(dt[s], 16, 32);
        ri[s] = __shfl_xor(it[s], 16, 32);
    }
    #pragma unroll
    for (int s = 0; s < KTOP; ++s) {
        float cz = rz[s], cd = rd[s];
        int   ci = ri[s];
        #pragma unroll
        for (int t2 = 0; t2 < KTOP; ++t2) {
            bool lt = cz < zt[t2];
            float tz = lt ? zt[t2] : cz;
            float td = lt ? dt[t2] : cd;
            int   ti = lt ? it[t2] : ci;
            zt[t2] = lt ? cz : zt[t2];
            dt[t2] = lt ? cd : dt[t2];
            it[t2] = lt ? ci : it[t2];
            cz = tz; cd = td; ci = ti;
        }
    }

    // ---- front-to-back compositing ----
    float trans = 1.0f, acc0 = 0.0f, acc1 = 0.0f, acc2 = 0.0f;
    float wsum = 0.0f, wzsum = 0.0f;
    #pragma unroll
    for (int s = 0; s < KTOP; ++s) {
        bool hit = zt[s] < 0.5f * ZFAR_;
        float w  = hit ? (1.0f - dt[s] * (1.0f / R2_)) : 0.0f;
        int  idx = hit ? it[s] : 0;
        float f0 = feats[3*idx+0], f1 = feats[3*idx+1], f2 = feats[3*idx+2];
        float cw = w * trans;
        acc0 += cw * f0; acc1 += cw * f1; acc2 += cw * f2;
        wsum  += w;
        wzsum += w * (hit ? zt[s] : 0.0f);
        trans *= (1.0f - w);
    }

    if (!hi) {
        const int pix = pj * W_ + pi;
        out[3*pix+0] = fminf(fmaxf(acc0, 0.0f), 1.0f);
        out[3*pix+1] = fminf(fmaxf(acc1, 0.0f), 1.0f);
        out[3*pix+2] = fminf(fmaxf(acc2, 0.0f), 1.0f);
        out[H_*W_*3 + pix] = 1.0f - trans;                    // mask
        out[H_*W_*4 + pix] = wzsum / fmaxf(wsum, EPS_);       // depth
    }
}

// ---------------- host-side launch ----------------
extern "C" void kernel_launch(void* const* d_in, const int* in_sizes, int n_in,
                              void* d_out, int out_size, void* d_ws, size_t ws_size,
                              hipStream_t stream)
{
    const float* pts       = (const float*)d_in[0];   // (P,3)
    const float* feats     = (const float*)d_in[1];   // (P,3)
    const float* Rm        = (const float*)d_in[2];   // (3,3)
    const float* Tv        = (const float*)d_in[3];   // (3,)
    const float* focal     = (const float*)d_in[4];   // (2,)
    const float* principal = (const float*)d_in[5];   // (2,)
    float* out = (float*)d_out;

    char* ws = (char*)d_ws;
    float4*   proj   = (float4*)(ws + WS_PROJ_);      // P * 16B
    int*      counts = (int*)(ws + WS_COUNTS_);       // 256 * 4
    int*      offs   = (int*)(ws + WS_OFFS_);         // 256 * 4
    int*      cursor = (int*)(ws + WS_CURSOR_);       // 256 * 4
    unsigned* list   = (unsigned*)(ws + WS_LIST_);    // 4*P * 4B
    float4*   dummy  = (float4*)(ws + WS_DUMMY_);     // 16B pad record

    hipMemsetAsync(counts, 0, 256 * sizeof(int), stream);

    const int blocks = (P_ + 255) / 256;
    proj_count_kernel<<<blocks, 256, 0, stream>>>(pts, Rm, Tv, focal, principal, proj, counts);
    scan_kernel<<<1, 256, 0, stream>>>(counts, offs, cursor, dummy);
    scatter_kernel<<<blocks, 256, 0, stream>>>(proj, cursor, list);
    render_kernel<<<NTX_*NTY_, 512, 0, stream>>>(proj, feats, offs, counts, list, out);
}